// Toymodel_60601988546596
// MI455X (gfx1250) — compile-verified
//
#include <hip/hip_runtime.h>
#include <hip/hip_bf16.h>
#include <stdint.h>

#define N_CAM     6
#define C_DIM     64
#define HW        180224      // 256*704
#define N_IDX     26000
#define OUT_COLS  160000      // 200*200*4
#define QTOT      40000       // 200*200
#define COLS_PER_BLK 128
#define Q_PER_BLK    32

typedef float  v4f __attribute__((ext_vector_type(4)));
typedef uint32_t v4u __attribute__((ext_vector_type(4)));

// ---------------------------------------------------------------------------
// Kernel 1: zero the winner-key array with wide b128 stores.
// ---------------------------------------------------------------------------
__global__ void k_init_winner(uint32_t* __restrict__ win) {
    int i = blockIdx.x * blockDim.x + threadIdx.x;     // over OUT_COLS/4
    if (i < OUT_COLS / 4) {
        v4u z = {0u, 0u, 0u, 0u};
        reinterpret_cast<v4u*>(win)[i] = z;
    }
}

// ---------------------------------------------------------------------------
// Kernel 2: scatter pass. Key = cam*N_IDX + j + 1 is monotone in camera
// order, so atomicMax implements "later camera overwrites earlier"; within a
// camera (JAX order unspecified for duplicates) the highest j wins,
// deterministically on every replay.
// ---------------------------------------------------------------------------
__global__ void k_scatter_winner(const int* __restrict__ scat,
                                 uint32_t* __restrict__ win) {
    int i = blockIdx.x * blockDim.x + threadIdx.x;
    if (i < N_CAM * N_IDX) {
        int s = scat[i];
        atomicMax(&win[s], (uint32_t)(i + 1));
    }
}

// ---------------------------------------------------------------------------
// Kernel 3 (fused resolve + gather + permute-store):
//   - decode 128 winner keys -> packed (cam<<24 | gather_g) in LDS
//   - async-gather the 128-col x 64-ch f32 tile into LDS via the CDNA5
//     GLOBAL_LOAD_ASYNC_TO_LDS_B32 path (ASYNCcnt, no VGPR landing zones)
//   - emit the final [4, C, 200*200] layout with coalesced non-temporal
//     b128 stores (512 B per wave32 per instruction).
// Every output element is written exactly once (zeros for unwritten cols).
// ---------------------------------------------------------------------------
__global__ void __launch_bounds__(256)
k_gather_out(const float* __restrict__ feat,
             const uint32_t* __restrict__ win,
             const int* __restrict__ gath,
             float* __restrict__ out) {
    __shared__ uint32_t rcache[COLS_PER_BLK];
    __shared__ float    tile[C_DIM * COLS_PER_BLK];   // 32 KB of 320 KB WGP LDS

    const int tid = threadIdx.x;
    const int s0  = blockIdx.x * COLS_PER_BLK;
    const int q0  = blockIdx.x * Q_PER_BLK;

    // Resolve stage: winner key -> (cam, gather_index) pack. g < 2^24.
    if (tid < COLS_PER_BLK) {
        uint32_t k = win[s0 + tid];
        uint32_t r = 0xFFFFFFFFu;
        if (k != 0u) {
            uint32_t idx = k - 1u;                  // cam*N_IDX + j
            uint32_t cam = idx / N_IDX;
            uint32_t g   = (uint32_t)gath[idx];
            r = (cam << 24) | g;
        }
        rcache[tid] = r;
    }
    __syncthreads();

    // Phase 1: each of the 8192 tile slots is touched by exactly ONE op:
    // either an async global->LDS gather (ASYNCcnt) or a ds zero-store
    // (DScnt). No slot mixes counters -> no DScnt/ASYNCcnt ordering hazard.
    #pragma unroll 4
    for (int i = 0; i < 32; ++i) {
        int k   = tid + i * 256;                    // 0..8191
        int col = k & (COLS_PER_BLK - 1);
        int c   = k >> 7;
        uint32_t r = rcache[col];
        int lidx = c * COLS_PER_BLK + col;
        if (r != 0xFFFFFFFFu) {
            uint32_t cam = r >> 24;
            uint32_t g   = r & 0x00FFFFFFu;
            const float* src = feat + ((size_t)(cam * C_DIM + c) * HW + g);
            uint64_t gaddr = (uint64_t)(uintptr_t)src;
            // Generic-pointer truncation: the LDS aperture keeps the
            // wave-relative LDS byte offset in addr[31:0].
            uint32_t laddr = (uint32_t)(uintptr_t)(&tile[lidx]);
            asm volatile("global_load_async_to_lds_b32 %0, %1, off"
                         :: "v"(laddr), "v"(gaddr)
                         : "memory");
        } else {
            tile[lidx] = 0.0f;
        }
    }
    // Each wave drains its own async gathers; the barrier (which drains
    // DScnt) then makes the whole tile visible to the workgroup.
    asm volatile("s_wait_asynccnt 0" ::: "memory");
    __syncthreads();

    // Phase 2: out[d*C*40000 + c*40000 + q] = out2[c, 4q+d].
    // One float4 per lane: 2048 stores/block, each wave writes 512 B of a
    // contiguous 128 B-per-(c,d) row group -> fully coalesced NT b128.
    #pragma unroll
    for (int i = 0; i < 8; ++i) {
        int e   = tid + i * 256;                    // 0..2047
        int lq4 = e & 7;                            // float4 index within row
        int c   = (e >> 3) & (C_DIM - 1);
        int d   = e >> 9;                           // 0..3
        int lqb = lq4 * 4;
        v4f v;
        v.x = tile[c * COLS_PER_BLK + (lqb + 0) * 4 + d];
        v.y = tile[c * COLS_PER_BLK + (lqb + 1) * 4 + d];
        v.z = tile[c * COLS_PER_BLK + (lqb + 2) * 4 + d];
        v.w = tile[c * COLS_PER_BLK + (lqb + 3) * 4 + d];
        size_t off = (size_t)d * (C_DIM * QTOT) + (size_t)c * QTOT + (q0 + lqb);
        __builtin_nontemporal_store(v, reinterpret_cast<v4f*>(out + off));
    }
}

// ---------------------------------------------------------------------------
extern "C" void kernel_launch(void* const* d_in, const int* in_sizes, int n_in,
                              void* d_out, int out_size, void* d_ws, size_t ws_size,
                              hipStream_t stream) {
    const float* feat = (const float*)d_in[0];     // [6, 64, 256, 704] f32
    const int*   gath = (const int*)d_in[1];       // [6, 26000]
    const int*   scat = (const int*)d_in[2];       // [6, 26000]
    float*       out  = (float*)d_out;             // [1, 256, 200, 200]

    uint32_t* win = (uint32_t*)d_ws;               // 160000 u32 (640 KB)

    k_init_winner   <<<(OUT_COLS / 4 + 255) / 256,  256, 0, stream>>>(win);
    k_scatter_winner<<<(N_CAM * N_IDX + 255) / 256, 256, 0, stream>>>(scat, win);
    k_gather_out    <<<OUT_COLS / COLS_PER_BLK,     256, 0, stream>>>(feat, win, gath, out);
}